// LSM_70652212019466
// MI455X (gfx1250) — compile-verified
//
#include <hip/hip_runtime.h>
#include <hip/hip_bf16.h>
#include <math.h>

// ---------------------------------------------------------------------------
// Latent-space model loss for MI455X (gfx1250, wave32).
//
// Dense term: sum_{i!=j} exp(-||Zs_i - Zs_j + eps||) over S=8192 sampled rows.
//   ||zi - zj + eps||^2 = u_i + v_j + D*eps^2 - 2 * (zi . zj)
//   The Gram matrix Zs @ Zs^T (8192x8192, K=8) is computed with
//   V_WMMA_F32_16X16X4_F32 (2 WMMAs per 16x16 tile), freeing the VALU for the
//   67M exp + 67M sqrt ops that dominate the runtime. sqrt/exp use the single-
//   instruction TRANS ops (v_sqrt_f32 / v_exp_f32) which co-execute with the
//   pk-FMA epilogue and WMMA issue.
// Sparse term: sum over 1M edges of -||z_i - z_j + eps|| + alpha.
// ---------------------------------------------------------------------------

#define S_SIZE   8192
#define DIM      8
#define EPS_F    1e-6f
#define JCHUNK   1024     // j-columns staged in LDS per block (32 KB)

typedef __attribute__((ext_vector_type(2))) float v2f;
typedef __attribute__((ext_vector_type(8))) float v8f;

__device__ __forceinline__ float fast_sqrtf(float x) {
    return __builtin_amdgcn_sqrtf(x);   // single v_sqrt_f32 (~1 ULP)
}

// ---------------------------------------------------------------------------
// Kernel 1: gather sampled rows, precompute u_i / v_j row statistics.
// ---------------------------------------------------------------------------
__global__ __launch_bounds__(256) void lsm_gather_kernel(
    const float* __restrict__ latent, const int* __restrict__ sidx,
    float* __restrict__ Zs, float* __restrict__ U, float* __restrict__ V)
{
    int t = blockIdx.x * blockDim.x + threadIdx.x;
    if (t >= S_SIZE) return;
    int idx = sidx[t];
    const float* src = latent + (size_t)idx * DIM;
    float ss = 0.f, ls = 0.f;
#pragma unroll
    for (int d = 0; d < DIM; ++d) {
        float z = src[d];
        Zs[t * DIM + d] = z;
        ss += z * z;
        ls += z;
    }
    U[t] = ss + 2.f * EPS_F * ls;   // row i contribution
    V[t] = ss - 2.f * EPS_F * ls;   // row j contribution
}

// ---------------------------------------------------------------------------
// Kernel 2: dense pairwise term via WMMA.
// grid = (512 i-tiles, 8 j-chunks); block = 256 threads = 8 wave32s.
// Each wave computes 8 independent 16x16 Gram tiles (2x v_wmma_f32_16x16x4_f32
// each, K=8), then per-lane does 8x { fma, sqrt, exp, add }.
// ---------------------------------------------------------------------------
__global__ __launch_bounds__(256) void lsm_dense_kernel(
    const float* __restrict__ Zs, const float* __restrict__ U,
    const float* __restrict__ V, float* __restrict__ partials)
{
    __shared__ float sB[JCHUNK * DIM];   // 32 KB: j-chunk of Zs
    __shared__ float sV[JCHUNK];         // 4 KB
    __shared__ float sU[16];
    __shared__ float sRed[256];

    const int tid   = threadIdx.x;
    const int iBase = blockIdx.x * 16;
    const int jBase = blockIdx.y * JCHUNK;

    // Stage B chunk + per-row stats into LDS (b128 copies).
    const float4* gB  = (const float4*)(Zs + (size_t)jBase * DIM);
    float4*       sB4 = (float4*)sB;
#pragma unroll
    for (int k = 0; k < (JCHUNK * DIM / 4) / 256; ++k)
        sB4[tid + k * 256] = gB[tid + k * 256];
#pragma unroll
    for (int k = 0; k < JCHUNK / 256; ++k)
        sV[tid + k * 256] = V[jBase + tid + k * 256];
    if (tid < 16) sU[tid] = U[iBase + tid];
    __syncthreads();

    const int wave = tid >> 5;
    const int lane = tid & 31;
    const int l    = lane & 15;   // M (A) / N (B) index
    const int h    = lane >> 4;   // lane-half selects K pair {0,1} vs {2,3}

    // A fragment (16x4 f32 layout): VGPR0 = K even, VGPR1 = K odd,
    // lane halves split K in {0,1}/{2,3}. Same rows for all 8 tiles.
    const float* aRow = Zs + (size_t)(iBase + l) * DIM;
    v2f a01, a23;
    a01.x = aRow[2 * h];     a01.y = aRow[2 * h + 1];      // K = 0..3
    a23.x = aRow[4 + 2 * h]; a23.y = aRow[5 + 2 * h];      // K = 4..7

    const float cEps = (float)DIM * EPS_F * EPS_F;
    float partial = 0.f;

    // 64 j-tiles in the chunk; wave w takes tiles w, w+8, ..., w+56.
    // Uniform per-wave trip count -> EXEC all-1s at every WMMA.
    for (int jt = wave; jt < JCHUNK / 16; jt += 8) {
        const float* bRow = sB + (size_t)(jt * 16 + l) * DIM;   // B(k,n)=Zs[j_n][k]
        v2f b01, b23;
        b01.x = bRow[2 * h];     b01.y = bRow[2 * h + 1];
        b23.x = bRow[4 + 2 * h]; b23.y = bRow[5 + 2 * h];

        v8f acc = {};
        acc = __builtin_amdgcn_wmma_f32_16x16x4_f32(
            false, a01, false, b01, (short)0, acc, false, false);
        acc = __builtin_amdgcn_wmma_f32_16x16x4_f32(
            false, a23, false, b23, (short)0, acc, false, false);

        const float vv = sV[jt * 16 + l];   // column stat (N = l)
#pragma unroll
        for (int m = 0; m < 8; ++m) {
            // D layout: VGPR m holds row M = m + 8*h, col N = l.
            float d2 = sU[m + 8 * h] + vv + cEps - 2.0f * acc[m];
            d2 = d2 > 0.f ? d2 : 0.f;
            partial += __expf(-fast_sqrtf(d2));
        }
    }

    sRed[tid] = partial;
    __syncthreads();
#pragma unroll
    for (int s = 128; s > 0; s >>= 1) {
        if (tid < s) sRed[tid] += sRed[tid + s];
        __syncthreads();
    }
    if (tid == 0) partials[blockIdx.y * gridDim.x + blockIdx.x] = sRed[0];
}

// ---------------------------------------------------------------------------
// Kernel 3: sparse positive-edge distances (grid-stride, b128 row loads).
// ---------------------------------------------------------------------------
__global__ __launch_bounds__(256) void lsm_edge_kernel(
    const float* __restrict__ latent, const int* __restrict__ ei,
    const int* __restrict__ ej, int nE, float* __restrict__ partials)
{
    __shared__ float sRed[256];
    float partial = 0.f;
    for (int e = blockIdx.x * blockDim.x + threadIdx.x; e < nE;
         e += gridDim.x * blockDim.x) {
        const float* a = latent + (size_t)ei[e] * DIM;
        const float* b = latent + (size_t)ej[e] * DIM;
        float s = 0.f;
#pragma unroll
        for (int d = 0; d < DIM; ++d) {
            float df = a[d] - b[d] + EPS_F;
            s += df * df;
        }
        partial += fast_sqrtf(s);
    }
    sRed[threadIdx.x] = partial;
    __syncthreads();
#pragma unroll
    for (int s = 128; s > 0; s >>= 1) {
        if (threadIdx.x < s) sRed[threadIdx.x] += sRed[threadIdx.x + s];
        __syncthreads();
    }
    if (threadIdx.x == 0) partials[blockIdx.x] = sRed[0];
}

// ---------------------------------------------------------------------------
// Kernel 4: deterministic final reduction + loss algebra.
// ---------------------------------------------------------------------------
__global__ __launch_bounds__(256) void lsm_final_kernel(
    const float* __restrict__ densePartials, int nD,
    const float* __restrict__ edgePartials, int nEp,
    const float* __restrict__ alphaPtr, float* __restrict__ out, int nEdges)
{
    __shared__ float sRed[256];

    float acc = 0.f;
    for (int i = threadIdx.x; i < nD; i += 256) acc += densePartials[i];
    sRed[threadIdx.x] = acc;
    __syncthreads();
#pragma unroll
    for (int s = 128; s > 0; s >>= 1) {
        if (threadIdx.x < s) sRed[threadIdx.x] += sRed[threadIdx.x + s];
        __syncthreads();
    }
    float denseSum = sRed[0];
    __syncthreads();

    acc = 0.f;
    for (int i = threadIdx.x; i < nEp; i += 256) acc += edgePartials[i];
    sRed[threadIdx.x] = acc;
    __syncthreads();
#pragma unroll
    for (int s = 128; s > 0; s >>= 1) {
        if (threadIdx.x < s) sRed[threadIdx.x] += sRed[threadIdx.x + s];
        __syncthreads();
    }
    float edgeSum = sRed[0];

    if (threadIdx.x == 0) {
        float alpha = alphaPtr[0];
        // Diagonal: dist(i,i) = sqrt(D)*eps exactly -> subtract analytically.
        float diag    = (float)S_SIZE * __expf(-fast_sqrtf((float)DIM) * EPS_F);
        float offdiag = denseSum - diag;
        float e1      = __expf(1.0f);
        float z1      = __expf(alpha) * 0.5f * (e1 * e1) * offdiag;
        float z2      = -edgeSum + (float)nEdges * alpha;
        out[0] = z2 - z1;
    }
}

// ---------------------------------------------------------------------------
// Launch. Workspace layout (floats):
//   [0,65536)        Zs gathered sample (8192 x 8)
//   [65536,73728)    U
//   [73728,81920)    V
//   [81920,86016)    dense partials (512*8 = 4096)
//   [86016,87040)    edge partials  (1024)
// ---------------------------------------------------------------------------
extern "C" void kernel_launch(void* const* d_in, const int* in_sizes, int n_in,
                              void* d_out, int out_size, void* d_ws, size_t ws_size,
                              hipStream_t stream)
{
    const float* latent = (const float*)d_in[0];
    const float* alpha  = (const float*)d_in[1];
    const int*   sidx   = (const int*)d_in[2];
    const int*   ei     = (const int*)d_in[3];
    const int*   ej     = (const int*)d_in[4];
    const int    nE     = in_sizes[3];

    float* ws = (float*)d_ws;
    float* Zs = ws;
    float* U  = ws + 65536;
    float* V  = ws + 73728;
    float* dP = ws + 81920;
    float* eP = ws + 86016;

    lsm_gather_kernel<<<S_SIZE / 256, 256, 0, stream>>>(latent, sidx, Zs, U, V);

    dim3 gDense(S_SIZE / 16, S_SIZE / JCHUNK);   // (512, 8)
    lsm_dense_kernel<<<gDense, 256, 0, stream>>>(Zs, U, V, dP);

    lsm_edge_kernel<<<1024, 256, 0, stream>>>(latent, ei, ej, nE, eP);

    lsm_final_kernel<<<1, 256, 0, stream>>>(dP, 512 * 8, eP, 1024, alpha,
                                            (float*)d_out, nE);
}